// LSTMModel_29815662969470
// MI455X (gfx1250) — compile-verified
//
#include <hip/hip_runtime.h>
#include <hip/hip_bf16.h>
#include <math.h>

// Problem dims
#define VOCAB  8000
#define EMBED  512
#define HIDDEN 1024
#define BATCH  32
#define SEQ    512
#define GATES4 4096   // 4*HIDDEN
#define TOKENS 16384  // BATCH*SEQ

typedef __attribute__((ext_vector_type(2))) float v2f;
typedef __attribute__((ext_vector_type(8))) float v8f;

// D = A(16x4 f32) * B(4x16 f32) + C(16x16 f32), wave32.
// A layout: lane L<16 holds A[L][k],A[L][k+1]; lane L>=16 holds A[L-16][k+2],A[L-16][k+3]
// B layout: lane L<16 holds B[k][n],B[k+1][n];  lane L>=16 holds B[k+2][n],B[k+3][n]
// C layout: vgpr r: lanes0-15 -> (M=r, N=lane); lanes16-31 -> (M=r+8, N=lane-16)
static __device__ __forceinline__ v8f wmma_f32(v2f a, v2f b, v8f c) {
  return __builtin_amdgcn_wmma_f32_16x16x4_f32(false, a, false, b, (short)0, c,
                                               false, false);
}

static __device__ __forceinline__ float sigmoidf_(float v) {
  return 1.0f / (1.0f + __expf(-v));
}

// CDNA5 async copy: global -> LDS, tracked by ASYNCcnt.
// The LDS byte address is derived from the REAL destination pointer via an
// addrspacecast to LDS + ptrtoint: the cast folds to the exact LDS offset,
// and the ptrtoint CAPTURES the shared tile so the "memory"-clobbering asm
// is treated as a potential writer of it (keeps the ds_load reads alive —
// a pure integer offset got them folded to undef in round 2). No "m"
// constraint: the AMDGPU backend cannot select inline-asm memory operands.
static __device__ __forceinline__ void async_ld_b128(void* lds_dst,
                                                     const void* gaddr) {
  __attribute__((address_space(3))) char* p3 =
      (__attribute__((address_space(3))) char*)lds_dst;
  unsigned lds_byte = (unsigned)(uintptr_t)p3;
  asm volatile("global_load_async_to_lds_b128 %0, %1, off"
               :
               : "v"(lds_byte), "v"((unsigned long long)(uintptr_t)gaddr)
               : "memory");
}
static __device__ __forceinline__ void wait_async0() {
  asm volatile("s_wait_asynccnt 0x0" ::: "memory");
}

// ---------------------------------------------------------------------------
// Kernel 0: zero h (double buffered), c, and the grid-barrier counter.
// ---------------------------------------------------------------------------
__global__ void init_state_kernel(float* __restrict__ h2, float* __restrict__ c,
                                  unsigned int* __restrict__ counter) {
  int i = blockIdx.x * blockDim.x + threadIdx.x;
  if (i < 2 * BATCH * HIDDEN) h2[i] = 0.0f;
  if (i < BATCH * HIDDEN)     c[i]  = 0.0f;
  if (i == 0)                 *counter = 0u;
}

// ---------------------------------------------------------------------------
// Kernel 1: x_gates[s][b][4H] = emb[x[b][s]] @ W_x + bias
// Block tile: 64 rows x 64 cols. 128 threads = 4 waves; each wave owns one
// 16-col slice and FOUR 16-row tiles -> every B fragment feeds 4 WMMAs.
// A staged via double-buffered async-to-LDS (64-wide K chunks).
// m = s*32 + b  (so x_gates row-major [S][B][4H] == [m][4H])
// ---------------------------------------------------------------------------
__global__ __launch_bounds__(128)
void xgates_kernel(const int* __restrict__ x, const float* __restrict__ emb,
                   const float* __restrict__ Wf, const float* __restrict__ Wi,
                   const float* __restrict__ Wg, const float* __restrict__ Wo,
                   const float* __restrict__ bf, const float* __restrict__ bi,
                   const float* __restrict__ bg, const float* __restrict__ bo,
                   float* __restrict__ xg) {
  __shared__ float As[2][64][68];  // 2 x (64 rows x 64 K) + 4 pad (16B-aligned rows)

  const int tid  = threadIdx.x;
  const int lane = tid & 31;
  const int wave = tid >> 5;
  const int m0   = blockIdx.x * 64;         // block row base (token-row m)
  const int nb   = blockIdx.y;              // 0..63
  const int n0   = nb * 64 + wave * 16;     // gate-column base [0,4096)
  const int gate = n0 >> 10;
  const int cl0  = n0 & 1023;
  const float* Wgate = (gate == 0) ? Wf : (gate == 1) ? Wi : (gate == 2) ? Wg : Wo;
  const float* bgate = (gate == 0) ? bf : (gate == 1) ? bi : (gate == 2) ? bg : bo;

  const int col  = cl0 + (lane & 15);
  const int koff = (lane >> 4) << 1;
  const int arow = lane & 15;

  // Staging: thread -> (row = tid/2, 32-col half = tid&1), 8 x b128 per chunk.
  const int sr = tid >> 1;
  const int sc = (tid & 1) * 32;
  const int sm = m0 + sr;
  const int stok = x[(sm & 31) * SEQ + (sm >> 5)];   // m = s*32+b -> b=m&31, s=m>>5
  const float* srow = emb + (size_t)stok * EMBED;

  auto issue = [&](int buf, int kb) {
    #pragma unroll
    for (int q = 0; q < 8; ++q)
      async_ld_b128((void*)&As[buf][sr][sc + q * 4], srow + kb + sc + q * 4);
  };

  v8f acc[4];
  #pragma unroll
  for (int t = 0; t < 4; ++t)
    acc[t] = (v8f){0.f, 0.f, 0.f, 0.f, 0.f, 0.f, 0.f, 0.f};

  issue(0, 0);
  int buf = 0;
  for (int kb = 0; kb < EMBED; kb += 64, buf ^= 1) {
    wait_async0();
    __syncthreads();                       // As[buf] ready for all waves
    if (kb + 64 < EMBED) issue(buf ^ 1, kb + 64);  // overlap DMA with compute

    #pragma unroll 4
    for (int k = 0; k < 64; k += 4) {
      const int kk = k + koff;
      v2f b;
      const float* bp = Wgate + (size_t)(kb + kk) * HIDDEN + col;
      b.x = bp[0];
      b.y = bp[HIDDEN];
      #pragma unroll
      for (int t = 0; t < 4; ++t) {
        v2f a;
        a.x = As[buf][t * 16 + arow][kk];
        a.y = As[buf][t * 16 + arow][kk + 1];
        acc[t] = wmma_f32(a, b, acc[t]);
      }
    }
    __syncthreads();                       // all waves done reading As[buf]
  }

  const float bias = bgate[col];
  #pragma unroll
  for (int t = 0; t < 4; ++t) {
    const int mbase = m0 + t * 16 + ((lane >> 4) << 3);
    #pragma unroll
    for (int r = 0; r < 8; ++r) {
      const int m = mbase + r;
      xg[(size_t)m * GATES4 + n0 + (lane & 15)] = acc[t][r] + bias;
    }
  }
}

// ---------------------------------------------------------------------------
// Kernel 2: persistent LSTM recurrence. 16 WGs x 128 threads = 64 waves.
// Wave w owns hidden columns [w*16, w*16+16) for ALL four gates, so the
// c/h update is wave-local. h double-buffered; one grid barrier per step.
// ---------------------------------------------------------------------------
__global__ __launch_bounds__(128)
void lstm_rec_kernel(const float* __restrict__ xg,
                     const float* __restrict__ Wf, const float* __restrict__ Wi,
                     const float* __restrict__ Wg, const float* __restrict__ Wo,
                     float* __restrict__ h2,    // [2][32][1024]
                     float* __restrict__ c,     // [32][1024]
                     float* __restrict__ hs,    // [S*B][1024], row m = s*32+b
                     unsigned int* __restrict__ counter) {
  const int tid  = threadIdx.x;
  const int lane = tid & 31;
  const int wave = tid >> 5;
  const int w    = blockIdx.x * 4 + wave;     // 0..63
  const int j0   = w * 16;
  const int jcol = j0 + (lane & 15);
  const int koff = (lane >> 4) << 1;

  const float* Wh[4] = {Wf + (size_t)EMBED * HIDDEN, Wi + (size_t)EMBED * HIDDEN,
                        Wg + (size_t)EMBED * HIDDEN, Wo + (size_t)EMBED * HIDDEN};

  unsigned int target = 0;

  for (int s = 0; s < SEQ; ++s) {
    const float* hin  = h2 + (size_t)(s & 1) * BATCH * HIDDEN;
    float*       hout = h2 + (size_t)((s + 1) & 1) * BATCH * HIDDEN;

    v8f acc[2][4];
    #pragma unroll
    for (int mtl = 0; mtl < 2; ++mtl)
      #pragma unroll
      for (int g = 0; g < 4; ++g)
        acc[mtl][g] = (v8f){0.f, 0.f, 0.f, 0.f, 0.f, 0.f, 0.f, 0.f};

    for (int k = 0; k < HIDDEN; k += 4) {
      const int kk = k + koff;
      const float2 t0 = *(const float2*)(hin + (size_t)(lane & 15) * HIDDEN + kk);
      const float2 t1 = *(const float2*)(hin + (size_t)(16 + (lane & 15)) * HIDDEN + kk);
      v2f a0, a1;
      a0.x = t0.x; a0.y = t0.y;
      a1.x = t1.x; a1.y = t1.y;
      #pragma unroll
      for (int g = 0; g < 4; ++g) {
        const float* bp = Wh[g] + (size_t)kk * HIDDEN + jcol;
        v2f b;
        b.x = bp[0];
        b.y = bp[HIDDEN];
        acc[0][g] = wmma_f32(a0, b, acc[0][g]);
        acc[1][g] = wmma_f32(a1, b, acc[1][g]);
      }
    }

    // Elementwise gate math + state update (wave-local for columns j0..j0+15).
    const float* xgs = xg + (size_t)s * BATCH * GATES4;
    #pragma unroll
    for (int mtl = 0; mtl < 2; ++mtl) {
      #pragma unroll
      for (int r = 0; r < 8; ++r) {
        const int bb = mtl * 16 + r + ((lane >> 4) << 3);
        const size_t xb = (size_t)bb * GATES4 + jcol;
        const float gf = sigmoidf_(acc[mtl][0][r] + xgs[xb]);
        const float gi = sigmoidf_(acc[mtl][1][r] + xgs[xb + HIDDEN]);
        const float gg = tanhf    (acc[mtl][2][r] + xgs[xb + 2 * HIDDEN]);
        const float go = sigmoidf_(acc[mtl][3][r] + xgs[xb + 3 * HIDDEN]);
        const size_t ci = (size_t)bb * HIDDEN + jcol;
        const float cnew = gf * c[ci] + gi * gg;
        const float hnew = go * tanhf(cnew);
        c[ci]    = cnew;
        hout[ci] = hnew;
        hs[((size_t)s * BATCH + bb) * HIDDEN + jcol] = hnew;
      }
    }

    // Device-wide barrier before anyone reads hout as next step's hin.
    __syncthreads();
    target += gridDim.x;
    if (tid == 0) {
      __threadfence();
      __hip_atomic_fetch_add(counter, 1u, __ATOMIC_ACQ_REL, __HIP_MEMORY_SCOPE_AGENT);
      while (__hip_atomic_load(counter, __ATOMIC_ACQUIRE, __HIP_MEMORY_SCOPE_AGENT) < target) {
        __builtin_amdgcn_s_sleep(1);
      }
    }
    __syncthreads();
    __threadfence();
  }
}

// ---------------------------------------------------------------------------
// Kernel 3: logits[b][s][v] = hs[m] @ W_out + b_out   (m = s*32+b)
// Block tile: 64 rows x 64 cols, 4 waves; each wave: 4 M-tiles x 1 N-tile,
// so each W_out fragment feeds 4 WMMAs. A staged via double-buffered
// async-to-LDS, overlapping DMA with the WMMA stream.
// ---------------------------------------------------------------------------
__global__ __launch_bounds__(128)
void logits_kernel(const float* __restrict__ hs, const float* __restrict__ W_out,
                   const float* __restrict__ b_out, float* __restrict__ out) {
  __shared__ float As[2][64][68];

  const int tid  = threadIdx.x;
  const int lane = tid & 31;
  const int wave = tid >> 5;
  const int m0   = blockIdx.x * 64;       // token-row base
  const int nb   = blockIdx.y;            // 0..124
  const int n0   = nb * 64 + wave * 16;   // 0..7984
  const int col  = n0 + (lane & 15);
  const int koff = (lane >> 4) << 1;
  const int arow = lane & 15;

  const int sr = tid >> 1;
  const int sc = (tid & 1) * 32;
  const float* srow = hs + (size_t)(m0 + sr) * HIDDEN;

  auto issue = [&](int buf, int kb) {
    #pragma unroll
    for (int q = 0; q < 8; ++q)
      async_ld_b128((void*)&As[buf][sr][sc + q * 4], srow + kb + sc + q * 4);
  };

  v8f acc[4];
  #pragma unroll
  for (int t = 0; t < 4; ++t)
    acc[t] = (v8f){0.f, 0.f, 0.f, 0.f, 0.f, 0.f, 0.f, 0.f};

  issue(0, 0);
  int buf = 0;
  for (int kb = 0; kb < HIDDEN; kb += 64, buf ^= 1) {
    wait_async0();
    __syncthreads();
    if (kb + 64 < HIDDEN) issue(buf ^ 1, kb + 64);

    #pragma unroll 4
    for (int k = 0; k < 64; k += 4) {
      const int kk = k + koff;
      v2f b;
      const float* bp = W_out + (size_t)(kb + kk) * VOCAB + col;
      b.x = bp[0];
      b.y = bp[VOCAB];
      #pragma unroll
      for (int t = 0; t < 4; ++t) {
        v2f a;
        a.x = As[buf][t * 16 + arow][kk];
        a.y = As[buf][t * 16 + arow][kk + 1];
        acc[t] = wmma_f32(a, b, acc[t]);
      }
    }
    __syncthreads();
  }

  const float bias = b_out[col];
  #pragma unroll
  for (int t = 0; t < 4; ++t) {
    const int mbase = m0 + t * 16 + ((lane >> 4) << 3);
    #pragma unroll
    for (int r = 0; r < 8; ++r) {
      const int m  = mbase + r;
      const int bb = m & 31, s = m >> 5;
      out[((size_t)bb * SEQ + s) * VOCAB + col] = acc[t][r] + bias;
    }
  }
}

// ---------------------------------------------------------------------------
// Host entry
// ---------------------------------------------------------------------------
extern "C" void kernel_launch(void* const* d_in, const int* in_sizes, int n_in,
                              void* d_out, int out_size, void* d_ws, size_t ws_size,
                              hipStream_t stream) {
  const int*   x     = (const int*)  d_in[0];
  const float* emb   = (const float*)d_in[1];
  const float* Wf    = (const float*)d_in[2];
  const float* bf    = (const float*)d_in[3];
  const float* Wi    = (const float*)d_in[4];
  const float* bi    = (const float*)d_in[5];
  const float* Wg    = (const float*)d_in[6];
  const float* bg    = (const float*)d_in[7];
  const float* Wo    = (const float*)d_in[8];
  const float* bo    = (const float*)d_in[9];
  const float* W_out = (const float*)d_in[10];
  const float* b_out = (const float*)d_in[11];
  float* out = (float*)d_out;

  // Workspace layout (bytes)
  char* ws = (char*)d_ws;
  const size_t XG_BYTES = (size_t)SEQ * BATCH * GATES4 * 4;   // 256 MB
  const size_t HS_BYTES = (size_t)TOKENS * HIDDEN * 4;        //  64 MB
  const size_t H_BYTES  = (size_t)2 * BATCH * HIDDEN * 4;     // 256 KB
  const size_t C_BYTES  = (size_t)BATCH * HIDDEN * 4;         // 128 KB
  float* xg  = (float*)(ws);
  float* hsb = (float*)(ws + XG_BYTES);
  float* h2  = (float*)(ws + XG_BYTES + HS_BYTES);
  float* cst = (float*)(ws + XG_BYTES + HS_BYTES + H_BYTES);
  unsigned int* counter =
      (unsigned int*)(ws + XG_BYTES + HS_BYTES + H_BYTES + C_BYTES);

  // 0) zero state
  init_state_kernel<<<(2 * BATCH * HIDDEN + 255) / 256, 256, 0, stream>>>(
      h2, cst, counter);

  // 1) input-side gate pre-activations (one big GEMM with embedding gather)
  xgates_kernel<<<dim3(TOKENS / 64, GATES4 / 64), 128, 0, stream>>>(
      x, emb, Wf, Wi, Wg, Wo, bf, bi, bg, bo, xg);

  // 2) sequential recurrence (persistent kernel, grid barrier per step)
  lstm_rec_kernel<<<dim3(16), 128, 0, stream>>>(xg, Wf, Wi, Wg, Wo, h2, cst, hsb,
                                                counter);

  // 3) output projection (dominant GEMM, 524 MB store)
  logits_kernel<<<dim3(TOKENS / 64, VOCAB / 64), 128, 0, stream>>>(
      hsb, W_out, b_out, out);
}